// Transactional_Extension_55
// MI455X (gfx1250) — compile-verified
//
#include <hip/hip_runtime.h>
#include <stdint.h>

// out[b, t, k, p, d] = x[b, max(0, t+k+p-7), d]
//   B=16, T=1024, D=64, Tout=1017, k,p in [0,8)
// Pure data movement: ~267MB stores, 4MB reads. Stage each block's 15-row
// source window into LDS via the CDNA5 Tensor Data Mover, then stream
// coalesced non-temporal b128 stores (write-once output; keep L2 for the
// 64x-reused input).

typedef unsigned int u32x4 __attribute__((ext_vector_type(4)));
typedef int          i32x8 __attribute__((ext_vector_type(8)));
typedef int          i32x4 __attribute__((ext_vector_type(4)));
typedef float        f32x4 __attribute__((ext_vector_type(4)));

#define T_IN     1024
#define T_OUT    1017
#define DD       64
#define KP       64      // 8*8 window rows per (b,t)
#define MAX_ROWS 15

__global__ __launch_bounds__(256)
void window_gather_tdm(const float* __restrict__ x, float* __restrict__ out) {
  __shared__ float tile[MAX_ROWS * DD];   // 3840 B staging window

  const int t  = blockIdx.x;              // 0..1016
  const int b  = blockIdx.y;              // 0..15
  const int lo = (t >= 7) ? (t - 7) : 0;  // first source row in window
  const int cnt = t + 7 - lo + 1;         // rows in window (8..15)

  if (threadIdx.x == 0) {
    // ---- Build Tensor DMA Descriptor (D#) and issue TDM load ----
    // tile[0:cnt, 0:64] = x[b, lo:lo+cnt, 0:64]
    const uint64_t gaddr = (uint64_t)(uintptr_t)(x + ((size_t)b * T_IN + lo) * DD);
    const uint32_t lds_addr = (uint32_t)(uintptr_t)(void*)&tile[0]; // low 32b = LDS offset

    u32x4 g0;
    g0[0] = 1u;                                   // count=1 (valid), user mode
    g0[1] = lds_addr;                             // lds_addr  [63:32]
    g0[2] = (uint32_t)gaddr;                      // global_addr[31:0]
    g0[3] = ((uint32_t)(gaddr >> 32) & 0x1FFFFFFu)// global_addr[56:32]
          | (2u << 30);                           // type=2 ("image")

    i32x8 g1;
    g1[0] = (int)(2u << 16);                      // data_size=2 (4B); no pad/iter/mcast
    g1[1] = (int)((uint32_t)DD << 16);            // tensor_dim0 = 64   (bits 79:48)
    g1[2] = (int)(((uint32_t)(T_IN - lo) & 0xFFFFu) << 16); // tensor_dim1 (bits 111:80)
    g1[3] = (int)((uint32_t)DD << 16);            // tile_dim0 = 64     (bits 127:112)
    g1[4] = cnt;                                  // tile_dim1 = rows   (bits 143:128)
    g1[5] = DD;                                   // tensor_dim0_stride = 64 elements
    g1[6] = 0;                                    // dim0_stride hi / dim1_stride lo
    g1[7] = 0;                                    // dim1_stride hi (unused, 2D tile)

    i32x4 gz  = {0, 0, 0, 0};                     // groups 2/3 unused (2D tensor)
    i32x8 gz8 = {0, 0, 0, 0, 0, 0, 0, 0};         // extra group (clang-23 6-arg form)

    __builtin_amdgcn_tensor_load_to_lds(g0, g1, gz, gz, gz8, /*cpol=*/0);
    __builtin_amdgcn_s_wait_tensorcnt(0);         // TDM complete -> LDS valid
  }
  __syncthreads();

  // ---- Stream out 64 rows x 64 floats = 1024 float4 per block ----
  f32x4* out4 = (f32x4*)out + ((size_t)b * T_OUT + t) * (KP * DD / 4);
  const f32x4* tile4 = (const f32x4*)tile;

#pragma unroll
  for (int i = 0; i < 4; ++i) {
    const int f  = (int)threadIdx.x + i * 256;    // 0..1023 (float4 index in block)
    const int kp = f >> 4;                        // 0..63  (k*8+p)
    const int d4 = f & 15;                        // 0..15  (float4 within row)
    const int k  = kp >> 3;
    const int p  = kp & 7;
    int src = t + k + p - 7;                      // clamped source row
    if (src < 0) src = 0;
    const int lrow = src - lo;                    // 0..cnt-1 in LDS window
    const f32x4 v = tile4[lrow * (DD / 4) + d4];  // ds_load_b128
    __builtin_nontemporal_store(v, &out4[f]);     // global_store_b128 (TH=NT)
  }
}

extern "C" void kernel_launch(void* const* d_in, const int* in_sizes, int n_in,
                              void* d_out, int out_size, void* d_ws, size_t ws_size,
                              hipStream_t stream) {
  const float* x = (const float*)d_in[0];
  float* out = (float*)d_out;
  dim3 grid(T_OUT, 16);
  window_gather_tdm<<<grid, 256, 0, stream>>>(x, out);
}